// EnhancedUserEmbedding_55301998903803
// MI455X (gfx1250) — compile-verified
//
#include <hip/hip_runtime.h>
#include <stdint.h>

// Problem constants (from the reference)
#define NUSERS 20000
#define EDIM   64
#define KTOP   6          // K+1 = 6 (top-6, drop the max)
#define ALPHA  0.7f
#define BLK    256
#define NWAVE  (BLK / 32)

#define ROW_BYTES   (NUSERS * 4)          // 80000
#define NCHUNK16    (ROW_BYTES / 16)      // 5000 x 16B chunks
#define FULL_ITERS  (NCHUNK16 / BLK)      // 19 full unrolled async issues
#define TAIL_CHUNKS (NCHUNK16 - FULL_ITERS * BLK)   // 136

// One workgroup per batch user: stream the 20000-float cooc row into LDS with
// the CDNA5 async-to-LDS engine (b128 per lane, immediate offsets), select
// top-6 (tie-break: lower index, matching jax.lax.top_k), softmax entries
// 1..5, blend embeddings.
__global__ __launch_bounds__(BLK)
void user_embed_topk_blend(const int* __restrict__ user_idx,
                           const float* __restrict__ emb,
                           const float* __restrict__ cooc,
                           float* __restrict__ out)
{
    __shared__ float s_row[NUSERS];      // 80000 B: one cooc row
    __shared__ float s_wv[NWAVE];        // per-wave argmax value
    __shared__ int   s_wei[NWAVE];       // per-wave argmax element index
    __shared__ int   s_wtid[NWAVE];      // per-wave winning thread
    __shared__ float s_tval[KTOP];       // block top-6 values (descending)
    __shared__ int   s_tidx[KTOP];       // block top-6 indices
    __shared__ int   s_wtid_bc;          // broadcast: winning thread this round

    const int b = blockIdx.x;
    const int t = threadIdx.x;
    const int u = user_idx[b];
    const float* rowp = cooc + (size_t)u * NUSERS;

    // ---- Stage row -> LDS via async DMA (ASYNCcnt path, gfx1250) -----------
    // Lane base addresses; per-issue displacement goes into the 24-bit
    // instruction offset, which hardware adds to BOTH the LDS and the global
    // address (ISA 10.x async pseudocode), so one base pair serves all issues.
    {
        const unsigned laddr =
            (unsigned)(uintptr_t)(&s_row[0]) + (unsigned)t * 16u; // low 32 = LDS offset
        const unsigned long long gaddr =
            (unsigned long long)(uintptr_t)rowp + (unsigned long long)t * 16ull;
#pragma unroll
        for (int k = 0; k < FULL_ITERS; ++k) {
            asm volatile("global_load_async_to_lds_b128 %0, %1, off offset:%c2"
                         :: "v"(laddr), "v"(gaddr), "i"(k * BLK * 16)
                         : "memory");
        }
        if (t < TAIL_CHUNKS) {
            asm volatile("global_load_async_to_lds_b128 %0, %1, off offset:%c2"
                         :: "v"(laddr), "v"(gaddr), "i"(FULL_ITERS * BLK * 16)
                         : "memory");
        }
        asm volatile("s_wait_asynccnt 0" ::: "memory");
    }
    __syncthreads();

    // ---- Per-thread sorted top-6, vectorized float4 scan of LDS ------------
    float lv[KTOP];
    int   li[KTOP];
#pragma unroll
    for (int k = 0; k < KTOP; ++k) { lv[k] = -__builtin_inff(); li[k] = 0x7fffffff; }

    const float4* row4 = reinterpret_cast<const float4*>(&s_row[0]);
    for (int c = t; c < NUSERS / 4; c += BLK) {      // 5000 float4 chunks
        float4 x4 = row4[c];
        float xs[4] = { x4.x, x4.y, x4.z, x4.w };
#pragma unroll
        for (int j = 0; j < 4; ++j) {
            float x = xs[j];
            int   i = 4 * c + j;
            bool better = (x > lv[KTOP - 1]) || (x == lv[KTOP - 1] && i < li[KTOP - 1]);
            if (better) {
                float cv = x; int ci = i;
#pragma unroll
                for (int k = KTOP - 1; k >= 0; --k) {
                    bool gt = (cv > lv[k]) || (cv == lv[k] && ci < li[k]);
                    if (gt) { float tv = lv[k]; int ti = li[k];
                              lv[k] = cv; li[k] = ci; cv = tv; ci = ti; }
                }
            }
        }
    }

    // ---- 6 rounds of block argmax (wave32 shuffle + LDS cross-wave) --------
    const int lane = t & 31;
    const int wid  = t >> 5;
    int p = 0;   // pointer into my sorted (descending) private list

    for (int r = 0; r < KTOP; ++r) {
        float v  = (p < KTOP) ? lv[p] : -__builtin_inff();
        int   ei = (p < KTOP) ? li[p] : 0x7fffffff;
        int   wt = t;
#pragma unroll
        for (int off = 16; off > 0; off >>= 1) {
            float ov  = __shfl_xor(v,  off, 32);
            int   oei = __shfl_xor(ei, off, 32);
            int   owt = __shfl_xor(wt, off, 32);
            if (ov > v || (ov == v && oei < ei)) { v = ov; ei = oei; wt = owt; }
        }
        if (lane == 0) { s_wv[wid] = v; s_wei[wid] = ei; s_wtid[wid] = wt; }
        __syncthreads();
        if (t == 0) {
            float bv = s_wv[0]; int bei = s_wei[0]; int bwt = s_wtid[0];
#pragma unroll
            for (int w2 = 1; w2 < NWAVE; ++w2) {
                float ov = s_wv[w2]; int oei = s_wei[w2];
                if (ov > bv || (ov == bv && oei < bei)) {
                    bv = ov; bei = oei; bwt = s_wtid[w2];
                }
            }
            s_tval[r] = bv; s_tidx[r] = bei; s_wtid_bc = bwt;
        }
        __syncthreads();
        if (t == s_wtid_bc) ++p;   // winner advances; next-round write to
                                   // s_wtid_bc is behind next round's barrier
    }

    // ---- Softmax over entries 1..5 + gather-blend, 64 lanes write output ---
    if (t < EDIM) {
        float m = s_tval[1];                 // max of the kept 5 (sorted desc)
        float w[KTOP - 1];
        float wsum = 0.0f;
#pragma unroll
        for (int k = 0; k < KTOP - 1; ++k) {
            w[k] = __expf(s_tval[k + 1] - m);
            wsum += w[k];
        }
        float scale = (1.0f - ALPHA) / wsum;
        float acc = ALPHA * emb[(size_t)u * EDIM + t];
#pragma unroll
        for (int k = 0; k < KTOP - 1; ++k) {
            acc += scale * w[k] * emb[(size_t)s_tidx[k + 1] * EDIM + t];
        }
        out[(size_t)b * EDIM + t] = acc;
    }
}

extern "C" void kernel_launch(void* const* d_in, const int* in_sizes, int n_in,
                              void* d_out, int out_size, void* d_ws, size_t ws_size,
                              hipStream_t stream) {
    const int*   user_idx = (const int*)d_in[0];    // (BATCH,)  int32
    const float* emb      = (const float*)d_in[1];  // (NUM_USERS, 64) f32
    const float* cooc     = (const float*)d_in[2];  // (NUM_USERS, NUM_USERS) f32
    float*       out      = (float*)d_out;          // (BATCH, 64) f32

    const int batch = in_sizes[0];                  // 4096
    user_embed_topk_blend<<<batch, BLK, 0, stream>>>(user_idx, emb, cooc, out);
}